// UnfoldNd_14559939133582
// MI455X (gfx1250) — compile-verified
//
#include <hip/hip_runtime.h>
#include <stdint.h>

// Unfold (im2col) 3x3 / pad 1 / stride 1 on (8,64,224,224) f32.
// Pure data movement (0 FLOPs): TDM (tensor_load_to_lds) stages a 34-row band
// of one input plane in LDS exactly once from HBM; threads then replicate it
// 9x with (ki,kj) shifts via coalesced non-temporal global_store_b128.
// Roofline: ~103 MB read + ~925 MB write ~= 1.03 GB @ 23.3 TB/s => ~44 us.

namespace {
constexpr int kB = 8;
constexpr int kC = 64;
constexpr int kH = 224;
constexpr int kW = 224;
constexpr int kTH = 32;                 // output rows per block (band)
constexpr int kBands = kH / kTH;        // 7
constexpr int kL = kH * kW;             // 50176
constexpr int kRowF4 = kW / 4;          // 56 float4 chunks per row
// kTH * kRowF4 = 1792 = 7 * 256  -> exactly 7 iterations per thread
}

typedef uint32_t u32x4 __attribute__((ext_vector_type(4)));
typedef uint32_t u32x8 __attribute__((ext_vector_type(8)));
typedef float    f32x4 __attribute__((ext_vector_type(4)));  // NT-store vector

__global__ __launch_bounds__(256) void unfold_tdm_kernel(
    const float* __restrict__ x, float* __restrict__ out) {
  // LDS holds rows (r0-1 .. r0+kTH) of one input plane: 34*224*4 = 30.5 KB
  __shared__ float smem[(kTH + 2) * kW];

  const int blk  = blockIdx.x;          // 0 .. B*C*kBands-1
  const int band = blk % kBands;
  const int bc   = blk / kBands;        // b*C + c
  const int r0   = band * kTH;

  // Global rows actually fetched (clamped at plane edges)
  const int rlo = (r0 == 0) ? 0 : (r0 - 1);
  const int rhi = (r0 + kTH >= kH) ? (kH - 1) : (r0 + kTH);
  const int nrows = rhi - rlo + 1;
  const int lds_row0 = rlo - (r0 - 1);  // 1 for top band, else 0

  const uint64_t gaddr =
      (uint64_t)(uintptr_t)(x + (size_t)bc * kL + (size_t)rlo * kW);
  const uint32_t lds_addr =
      (uint32_t)(uintptr_t)(&smem[0]) + (uint32_t)(lds_row0 * kW * 4);

  // ---- Tensor Data Mover: one DMA per block, issued by wave 0 only
  // (TDM ignores EXEC and is per-wave, so gate on the wave, not the lane).
  if (threadIdx.x < 32) {
    // D# group 0 (ISA 8.3): count=1 | lds_addr | global_addr[56:0] | type=2
    u32x4 g0;
    g0.x = 1u;
    g0.y = lds_addr;
    g0.z = (uint32_t)gaddr;
    g0.w = ((uint32_t)(gaddr >> 32) & 0x01FFFFFFu) | (2u << 30);

    // D# group 1 (ISA 8.4): data_size=2 (4B), 2-D tile nrows x 224,
    // tensor_dim0=224, tensor_dim1=nrows, dim0_stride=224 elements.
    u32x8 g1;
    g1.s0 = 2u << 16;                       // workgroup_mask=0, data_size=4B
    g1.s1 = ((uint32_t)kW) << 16;           // tensor_dim0[15:0] @ bits[63:48]
    g1.s2 = ((uint32_t)nrows) << 16;        // dim0 hi=0 | tensor_dim1[15:0]
    g1.s3 = ((uint32_t)kW) << 16;           // dim1 hi=0 | tile_dim0=224
    g1.s4 = (uint32_t)nrows;                // tile_dim1=nrows, tile_dim2=0
    g1.s5 = (uint32_t)kW;                   // tensor_dim0_stride lo32
    g1.s6 = 0u;                             // stride hi | dim1_stride lo
    g1.s7 = 0u;                             // dim1_stride hi

    asm volatile("tensor_load_to_lds %0, %1"
                 :: "s"(g0), "s"(g1)
                 : "memory");
    __builtin_amdgcn_s_wait_tensorcnt(0);
  }

  // Zero-fill padding halo rows at image top/bottom (disjoint from DMA rows,
  // so this overlaps the in-flight TDM safely).
  if (r0 == 0)
    for (int j = threadIdx.x; j < kW; j += 256) smem[j] = 0.0f;
  if (r0 + kTH >= kH)
    for (int j = threadIdx.x; j < kW; j += 256) smem[(kTH + 1) * kW + j] = 0.0f;

  __syncthreads();

  // ---- Replicate band into 9 shifted output planes ----
  // out[b, c*9 + (ki*3+kj), i*224 + j] = x[b, c, i+ki-1, j+kj-1] (0 padded)
  const size_t out_bc = (size_t)bc * 9u * (size_t)kL;

#pragma unroll
  for (int it = 0; it < (kTH * kRowF4) / 256; ++it) {
    const int p     = (int)threadIdx.x + it * 256;
    const int i_loc = p / kRowF4;       // 0..kTH-1
    const int j     = (p % kRowF4) * 4; // 0,4,...,220
    const int i     = r0 + i_loc;

    // Gather 3 LDS rows x 6 columns (j-1 .. j+4); pad columns in registers.
    float v[3][6];
#pragma unroll
    for (int ki = 0; ki < 3; ++ki) {
      const float* row = &smem[(i_loc + ki) * kW]; // lds row = i - r0 + ki
      v[ki][0] = (j == 0) ? 0.0f : row[j - 1];
      v[ki][1] = row[j + 0];
      v[ki][2] = row[j + 1];
      v[ki][3] = row[j + 2];
      v[ki][4] = row[j + 3];
      v[ki][5] = (j + 4 >= kW) ? 0.0f : row[j + 4];
    }

    const size_t obase = out_bc + (size_t)i * kW + (size_t)j;
#pragma unroll
    for (int ki = 0; ki < 3; ++ki) {
#pragma unroll
      for (int kj = 0; kj < 3; ++kj) {
        f32x4 o;
        o.x = v[ki][kj + 0];
        o.y = v[ki][kj + 1];
        o.z = v[ki][kj + 2];
        o.w = v[ki][kj + 3];
        f32x4* dst =
            (f32x4*)(out + obase + (size_t)(ki * 3 + kj) * (size_t)kL);
        __builtin_nontemporal_store(o, dst);  // streaming write-once output
      }
    }
  }
}

extern "C" void kernel_launch(void* const* d_in, const int* in_sizes, int n_in,
                              void* d_out, int out_size, void* d_ws,
                              size_t ws_size, hipStream_t stream) {
  const float* x = (const float*)d_in[0];
  float* out = (float*)d_out;
  (void)in_sizes; (void)n_in; (void)out_size; (void)d_ws; (void)ws_size;

  dim3 grid(kB * kC * kBands);  // 8*64*7 = 3584 blocks
  dim3 block(256);              // 8 waves (wave32)
  hipLaunchKernelGGL(unfold_tdm_kernel, grid, block, 0, stream, x, out);
}